// HypernymVisual_acc_35055523070022
// MI455X (gfx1250) — compile-verified
//
#include <hip/hip_runtime.h>
#include <math.h>
#include <stdint.h>

// Problem constants (fixed by setup_inputs): B=512, D_vis=4096, E=1024, N=512
#define BB   512
#define DV   4096
#define EE   1024
#define NN   512
#define OUTC (NN + 1)   // 513 columns: [p_score, n_scores...]
#define KC   64         // K-chunk staged in LDS per pipeline stage

typedef __attribute__((ext_vector_type(2))) float v2f;
typedef __attribute__((ext_vector_type(8))) float v8f;
typedef int hv_v4i __attribute__((__vector_size__(16)));   // matches builtin proto

// ---------------------------------------------------------------------------
// Async global->LDS copy (CDNA5 ASYNCcnt path), with synchronous fallback.
// Probe-learned prototype: b128 variant takes (AS1 v4i* src, AS3 v4i* dst,
// imm offset, imm cpol).
// ---------------------------------------------------------------------------
#if __has_builtin(__builtin_amdgcn_global_load_async_to_lds_b128) && \
    __has_builtin(__builtin_amdgcn_s_wait_asynccnt)
#define HV_ASYNC 1
#endif

static __device__ __forceinline__ void hv_cp16(const void* g, void* l)
{
#if defined(HV_ASYNC)
    // 16B global -> LDS, no VGPR round trip; tracked by ASYNCcnt.
    __builtin_amdgcn_global_load_async_to_lds_b128(
        (__attribute__((address_space(1))) hv_v4i*)(unsigned long long)(uintptr_t)g,
        (__attribute__((address_space(3))) hv_v4i*)(unsigned)(uintptr_t)l,
        0, 0);
#else
    *(float4*)l = *(const float4*)g;
#endif
}

static __device__ __forceinline__ void hv_wait_async()
{
#if defined(HV_ASYNC)
    __builtin_amdgcn_s_wait_asynccnt(0);
#endif
}

// ---------------------------------------------------------------------------
// Kernel 1: vf_emb[B,E] = vf[B,DV] @ W[E,DV]^T + b[E]
// Block = 8 waves = 128(M) x 64(N) tile. K staged in LDS in 64-wide chunks,
// double-buffered with async global->LDS loads overlapping WMMA compute.
// Per wave / K-step(4): 1 A ds_load_b64 + 4 B ds_load_b64 + 4 wmma_f32_16x16x4.
// LDS row stride 68 floats: 16B-aligned rows, conflict-free b64 fragment reads
// (banks 4r+k .. 4r+k+3 cover all 64 banks across the wave).
// Grid: 4 m-blocks x 16 n-blocks = 64 blocks.
// ---------------------------------------------------------------------------
__global__ __launch_bounds__(256) void hv_gemm_bias_wmma(
    const float* __restrict__ vf, const float* __restrict__ W,
    const float* __restrict__ bias, float* __restrict__ emb)
{
    __shared__ float sa[2][128][68];
    __shared__ float sb[2][64][68];

    const int t    = threadIdx.x;
    const int lane = t & 31;
    const int wave = t >> 5;
    const int mb   = blockIdx.x & 3;     // m-block (128 rows)
    const int nb   = blockIdx.x >> 2;    // n-block (64 cols)
    const int m0b  = mb * 128;
    const int e0b  = nb * 64;

    const int r     = lane & 15;          // fragment row (A) / col (B)
    const int khalf = (lane >> 4) * 2;    // lanes 16-31 own K=2,3 of each step

    // --- staging assignment: A = 2048 float4 (8/thread), B = 1024 float4 (4/thread)
    // position p: row = p>>4, col4 = (p&15)*4; stride 256 keeps (p&15) fixed.
    const int col4 = (t & 15) * 4;

    // accumulators: C layout vgpr i: M = i + 8*(lane>>4), N = lane&15
    v8f acc[4];
#pragma unroll
    for (int j = 0; j < 4; ++j) {
        const float bv = bias[e0b + 16 * j + r];
#pragma unroll
        for (int i = 0; i < 8; ++i) acc[j][i] = bv;
    }

    // --- stage chunk kc into buffer buf
    auto stage = [&](int buf, int kc) {
#pragma unroll
        for (int i = 0; i < 8; ++i) {
            const int row = (t + i * 256) >> 4;   // 0..127
            hv_cp16(&vf[(size_t)(m0b + row) * DV + kc + col4],
                    &sa[buf][row][col4]);
        }
#pragma unroll
        for (int i = 0; i < 4; ++i) {
            const int row = (t + i * 256) >> 4;   // 0..63
            hv_cp16(&W[(size_t)(e0b + row) * DV + kc + col4],
                    &sb[buf][row][col4]);
        }
    };

    stage(0, 0);

    for (int kc = 0; kc < DV; kc += KC) {
        const int buf = (kc / KC) & 1;
        hv_wait_async();
        __syncthreads();                       // chunk `buf` visible to all waves
        if (kc + KC < DV) stage(buf ^ 1, kc + KC);  // overlap next chunk with compute

        const float* Ab = &sa[buf][wave * 16 + r][khalf];
#pragma unroll 4
        for (int k = 0; k < KC; k += 4) {
            const v2f a = *(const v2f*)(Ab + k);
#pragma unroll
            for (int j = 0; j < 4; ++j) {
                const v2f bf = *(const v2f*)&sb[buf][16 * j + r][k + khalf];
                acc[j] = __builtin_amdgcn_wmma_f32_16x16x4_f32(
                    false, a, false, bf, (short)0, acc[j], false, false);
            }
        }
    }

    const int m0  = m0b + wave * 16;
    const int mhi = (lane >> 4) * 8;
#pragma unroll
    for (int j = 0; j < 4; ++j)
#pragma unroll
        for (int i = 0; i < 8; ++i)
            emb[(size_t)(m0 + mhi + i) * EE + e0b + 16 * j + r] = acc[j][i];
}

// ---------------------------------------------------------------------------
// Kernel 2: n_scores[b][n] = -sqrt( sum_e relu(n_lfs[n][e] - emb[b][e])^2 )
// 32x32 output tile per 256-thread block; 32x64 operand slabs async-staged in LDS.
// ---------------------------------------------------------------------------
__global__ __launch_bounds__(256) void hv_nscores(
    const float* __restrict__ emb, const float* __restrict__ nl,
    float* __restrict__ out)
{
    __shared__ float se[32][68];
    __shared__ float sn[32][68];

    const int t  = threadIdx.x;
    const int b0 = (blockIdx.x & 15) * 32;
    const int n0 = (blockIdx.x >> 4) * 32;

    const int r  = t >> 3;          // output row within tile (0..31)
    const int c0 = (t & 7) * 4;     // output col base within tile

    float a0 = 0.f, a1 = 0.f, a2 = 0.f, a3 = 0.f;

    for (int ebase = 0; ebase < EE; ebase += 64) {
        __syncthreads();            // previous chunk fully consumed
        // 512 float4 per array, 2 per thread per array
        for (int p = t; p < 512; p += 256) {
            const int row  = p >> 4;
            const int col4 = (p & 15) * 4;
            hv_cp16(&emb[(size_t)(b0 + row) * EE + ebase + col4], &se[row][col4]);
            hv_cp16(&nl[(size_t)(n0 + row) * EE + ebase + col4], &sn[row][col4]);
        }
        hv_wait_async();
        __syncthreads();

#pragma unroll 8
        for (int e = 0; e < 64; ++e) {
            const float ev = se[r][e];
            float d;
            d = sn[c0 + 0][e] - ev; d = fmaxf(d, 0.f); a0 = fmaf(d, d, a0);
            d = sn[c0 + 1][e] - ev; d = fmaxf(d, 0.f); a1 = fmaf(d, d, a1);
            d = sn[c0 + 2][e] - ev; d = fmaxf(d, 0.f); a2 = fmaf(d, d, a2);
            d = sn[c0 + 3][e] - ev; d = fmaxf(d, 0.f); a3 = fmaf(d, d, a3);
        }
    }

    float* o = out + (size_t)(b0 + r) * OUTC + 1 + n0 + c0;
    o[0] = -sqrtf(a0);
    o[1] = -sqrtf(a1);
    o[2] = -sqrtf(a2);
    o[3] = -sqrtf(a3);
}

// ---------------------------------------------------------------------------
// Kernel 3: p_scores[b] = -sqrt( sum_e relu(p_lfs[b][e] - emb[b][e])^2 )
// One wave32 per row; float4 strided loads + shuffle reduction.
// ---------------------------------------------------------------------------
__global__ __launch_bounds__(256) void hv_pscores(
    const float* __restrict__ emb, const float* __restrict__ pl,
    float* __restrict__ out)
{
    const int lane = threadIdx.x & 31;
    const int b    = (blockIdx.x * 256 + threadIdx.x) >> 5;  // 0..511

    float acc = 0.f;
    for (int e = lane * 4; e < EE; e += 32 * 4) {
        const float4 pv = *(const float4*)&pl[(size_t)b * EE + e];
        const float4 ev = *(const float4*)&emb[(size_t)b * EE + e];
        float d;
        d = pv.x - ev.x; d = fmaxf(d, 0.f); acc = fmaf(d, d, acc);
        d = pv.y - ev.y; d = fmaxf(d, 0.f); acc = fmaf(d, d, acc);
        d = pv.z - ev.z; d = fmaxf(d, 0.f); acc = fmaf(d, d, acc);
        d = pv.w - ev.w; d = fmaxf(d, 0.f); acc = fmaf(d, d, acc);
    }
#pragma unroll
    for (int off = 16; off >= 1; off >>= 1)
        acc += __shfl_xor(acc, off, 32);
    if (lane == 0) out[(size_t)b * OUTC] = -sqrtf(acc);
}

// ---------------------------------------------------------------------------
extern "C" void kernel_launch(void* const* d_in, const int* in_sizes, int n_in,
                              void* d_out, int out_size, void* d_ws, size_t ws_size,
                              hipStream_t stream)
{
    const float* vf    = (const float*)d_in[0];  // [512, 4096]
    const float* p_lfs = (const float*)d_in[1];  // [512, 1024]
    const float* n_lfs = (const float*)d_in[2];  // [512, 1024]
    const float* W     = (const float*)d_in[3];  // [1024, 4096]
    const float* bias  = (const float*)d_in[4];  // [1024]
    float*       out   = (float*)d_out;          // [512, 513]
    float*       emb   = (float*)d_ws;           // [512, 1024] f32 scratch (2 MB)

    hv_gemm_bias_wmma<<<64, 256, 0, stream>>>(vf, W, bias, emb);
    hv_nscores<<<256, 256, 0, stream>>>(emb, n_lfs, out);
    hv_pscores<<<64, 256, 0, stream>>>(emb, p_lfs, out);
}